// GConvGRULinkPredictor_59296318488654
// MI455X (gfx1250) — compile-verified
//
#include <hip/hip_runtime.h>
#include <hip/hip_bf16.h>

typedef __attribute__((ext_vector_type(2))) float v2f;
typedef __attribute__((ext_vector_type(8))) float v8f;

// Problem constants (match reference)
constexpr int Bc = 8, Tt = 16, Nn = 512, Cc = 64, Hh = 64, Kk = 3, Ll = 2, Ee = 16384;
constexpr int BNC = Bc * Nn * Cc;  // 262144 (== BNH since C==H)
constexpr int BNH = Bc * Nn * Hh;  // 262144

// ---------------- utility kernels ----------------

__global__ __launch_bounds__(256) void k_zero(float* __restrict__ p, int n) {
  int i = blockIdx.x * 256 + threadIdx.x;
  if (i < n) p[i] = 0.0f;
}

__global__ __launch_bounds__(256) void k_edge_prep(const long long* __restrict__ ei,
                                                   int* __restrict__ src,
                                                   int* __restrict__ dst) {
  int i = blockIdx.x * 256 + threadIdx.x;  // exactly E threads
  src[i] = (int)ei[i];
  dst[i] = (int)ei[Ee + i];
}

__global__ __launch_bounds__(256) void k_deg(const int* __restrict__ src,
                                             const float* __restrict__ w,
                                             float* __restrict__ deg) {
  int i = blockIdx.x * 256 + threadIdx.x;  // exactly E threads
  atomicAdd(&deg[src[i]], w[i]);
}

__global__ __launch_bounds__(256) void k_norm(const int* __restrict__ src,
                                              const int* __restrict__ dst,
                                              const float* __restrict__ w,
                                              const float* __restrict__ deg,
                                              float* __restrict__ nw) {
  int i = blockIdx.x * 256 + threadIdx.x;  // exactly E threads
  float ds = deg[src[i]];
  float dd = deg[dst[i]];
  float is = ds > 0.0f ? rsqrtf(fmaxf(ds, 1e-12f)) : 0.0f;
  float id = dd > 0.0f ? rsqrtf(fmaxf(dd, 1e-12f)) : 0.0f;
  nw[i] = -w[i] * is * id;
}

// ---------------- CSR construction (counting sort by dst) ----------------

__global__ __launch_bounds__(256) void k_count(const int* __restrict__ dst,
                                               int* __restrict__ cnt) {
  int e = blockIdx.x * 256 + threadIdx.x;  // exactly E threads
  atomicAdd(&cnt[dst[e]], 1);
}

__global__ void k_scan(const int* __restrict__ cnt, int* __restrict__ rowptr) {
  if (blockIdx.x == 0 && threadIdx.x == 0) {
    int s = 0;
    for (int n = 0; n < Nn; ++n) { rowptr[n] = s; s += cnt[n]; }
    rowptr[Nn] = s;
  }
}

__global__ __launch_bounds__(256) void k_scatter(const int* __restrict__ src,
                                                 const int* __restrict__ dst,
                                                 const float* __restrict__ nw,
                                                 const int* __restrict__ rowptr,
                                                 int* __restrict__ fill,
                                                 int* __restrict__ srcS,
                                                 float* __restrict__ nwS) {
  int e = blockIdx.x * 256 + threadIdx.x;  // exactly E threads
  int d = dst[e];
  int pos = rowptr[d] + atomicAdd(&fill[d], 1);
  srcS[pos] = src[e];
  nwS[pos] = nw[e];
}

// ---------------- timestep slice ----------------

// xt[b,n,c] = x[b,t,n,c]
__global__ __launch_bounds__(256) void k_gather_xt(const float* __restrict__ x,
                                                   float* __restrict__ xt, int t) {
  int i = blockIdx.x * 256 + threadIdx.x;  // exactly B*N*C threads
  int b  = i >> 15;                        // N*C = 32768 = 2^15
  int nc = i & 32767;
  xt[i] = x[(((b * Tt) + t) << 15) + nc];
}

// ---------------- CSR gather spmv: out = scale * (L_hat @ in) + gamma * add ----------------
// One thread per (b, node, c); a wave covers 32 contiguous channels of one (b,node):
// uniform trip count per wave, coalesced in[] loads, no atomics.
__global__ __launch_bounds__(256) void k_spmv_csr(float* __restrict__ outp,
                                                  const float* __restrict__ inp,
                                                  const float* __restrict__ addp,
                                                  float gamma, float scale,
                                                  const int* __restrict__ rowptr,
                                                  const int* __restrict__ srcS,
                                                  const float* __restrict__ nwS) {
  int i = blockIdx.x * 256 + threadIdx.x;  // exactly B*N*64 threads
  int c    = i & 63;
  int node = (i >> 6) & (Nn - 1);
  int b    = i >> 15;
  int p0 = rowptr[node], p1 = rowptr[node + 1];
  float acc = 0.0f;
  for (int p = p0; p < p1; ++p)
    acc += nwS[p] * inp[(((b << 9) + srcS[p]) << 6) + c];
  outp[i] = scale * acc + gamma * addp[i];
}

// ---------------- gate activation kernels ----------------

__device__ __forceinline__ float sigmf(float x) { return 1.0f / (1.0f + expf(-x)); }

// z = sigmoid(preZ); r = sigmoid(preR); hr = h*r; z stored into preZ (biases folded into gemm)
__global__ __launch_bounds__(256) void k_gate_zr(float* __restrict__ preZ,
                                                 const float* __restrict__ preR,
                                                 const float* __restrict__ h,
                                                 float* __restrict__ hr) {
  int i = blockIdx.x * 256 + threadIdx.x;  // exactly B*N*H threads
  float z = sigmf(preZ[i]);
  float r = sigmf(preR[i]);
  preZ[i] = z;
  hr[i] = h[i] * r;
}

// h = z*h + (1-z)*tanh(preH)
__global__ __launch_bounds__(256) void k_gate_h(float* __restrict__ h,
                                                const float* __restrict__ z,
                                                const float* __restrict__ preH) {
  int i = blockIdx.x * 256 + threadIdx.x;  // exactly B*N*H threads
  float ht = tanhf(preH[i]);
  float zv = z[i];
  h[i] = zv * h[i] + (1.0f - zv) * ht;
}

// ---------------- WMMA f32 GEMM building block ----------------
// One 16x16 tile per wave; v_wmma_f32_16x16x4_f32, K=64 -> 16 WMMA per chain.
// A layout (16x4 f32, ISA 7.12.2): lanes 0-15 K={k,k+1}, lanes 16-31 K={k+2,k+3}.
// B layout (4x16): lane = column, VGPR pair = two K rows per lane half.
// C/D layout: VGPR v, lanes 0-15 -> M=v, lanes 16-31 -> M=v+8.
__device__ __forceinline__ v8f gemm_chain(v8f c, const float* __restrict__ A,
                                          const float* __restrict__ W,
                                          int row, int col, int half) {
  const float* Ar = A + row * 64 + 2 * half;
  const float* Wc = W + (2 * half) * 64 + col;
#pragma unroll
  for (int k = 0; k < 64; k += 4) {
    v2f a, b;
    a.x = Ar[k];
    a.y = Ar[k + 1];
    b.x = Wc[k * 64];
    b.y = Wc[k * 64 + 64];
    c = __builtin_amdgcn_wmma_f32_16x16x4_f32(false, a, false, b, (short)0, c,
                                              false, false);
  }
  return c;
}

// Fused gate pre-activation: Out = bias + sum_{k<3} Ax_k @ Wx[k] + Ah_k @ Wh[k]
// 96 WMMA ops per wave; accumulator never leaves VGPRs. Grid: 128 blocks x 256.
__global__ __launch_bounds__(256) void k_gemm6(
    const float* __restrict__ A0, const float* __restrict__ A1,
    const float* __restrict__ A2, const float* __restrict__ Wxg,
    const float* __restrict__ A3, const float* __restrict__ A4,
    const float* __restrict__ A5, const float* __restrict__ Whg,
    const float* __restrict__ bxg, const float* __restrict__ bhg,
    float* __restrict__ Out) {
  const int wave   = blockIdx.x * 8 + (threadIdx.x >> 5);
  const int lane   = threadIdx.x & 31;
  const int tile_n = wave & 3;
  const int tile_m = wave >> 2;
  const int half   = lane >> 4;
  const int l16    = lane & 15;
  const int row    = tile_m * 16 + l16;
  const int col    = tile_n * 16 + l16;

  float bias = bxg[col] + bhg[col];
  v8f c;
#pragma unroll
  for (int v = 0; v < 8; ++v) c[v] = bias;

  const float* As[6] = {A0, A1, A2, A3, A4, A5};
  const float* Ws[6] = {Wxg, Wxg + 4096, Wxg + 8192, Whg, Whg + 4096, Whg + 8192};
#pragma unroll
  for (int m = 0; m < 6; ++m) c = gemm_chain(c, As[m], Ws[m], row, col, half);

#pragma unroll
  for (int v = 0; v < 8; ++v)
    Out[(tile_m * 16 + v + 8 * half) * 64 + col] = c[v];
}

// Single matmul, write mode (no accumulator read): Out = A @ W
__global__ __launch_bounds__(256) void k_gemm1(const float* __restrict__ A,
                                               const float* __restrict__ W,
                                               float* __restrict__ Out) {
  const int wave   = blockIdx.x * 8 + (threadIdx.x >> 5);
  const int lane   = threadIdx.x & 31;
  const int tile_n = wave & 3;
  const int tile_m = wave >> 2;
  const int half   = lane >> 4;
  const int l16    = lane & 15;
  const int row    = tile_m * 16 + l16;
  const int col    = tile_n * 16 + l16;

  v8f c = {};
  c = gemm_chain(c, A, W, row, col, half);
#pragma unroll
  for (int v = 0; v < 8; ++v)
    Out[(tile_m * 16 + v + 8 * half) * 64 + col] = c[v];
}

// ---------------- all-pairs link predictor ----------------
// logits[b,i,j] = sum_h relu(sp[b,i,h] + tp[b,j,h] + b1[h]) * W2[h] + b2
__global__ __launch_bounds__(256) void k_pred(const float* __restrict__ sp,
                                              const float* __restrict__ tp,
                                              const float* __restrict__ b1,
                                              const float* __restrict__ W2,
                                              const float* __restrict__ b2,
                                              float* __restrict__ out) {
  __shared__ float ssh[64];
  __shared__ float w2sh[64];
  int bi = blockIdx.x;  // b*N + i
  int b  = bi >> 9;
  if (threadIdx.x < 64) {
    ssh[threadIdx.x]  = sp[bi * 64 + threadIdx.x] + b1[threadIdx.x];
    w2sh[threadIdx.x] = W2[threadIdx.x];
  }
  __syncthreads();
  float bias2 = b2[0];
  for (int j = threadIdx.x; j < Nn; j += 256) {
    const float* trow = tp + ((b << 9) + j) * 64;
    float acc = bias2;
#pragma unroll
    for (int hh = 0; hh < 64; ++hh)
      acc += fmaxf(ssh[hh] + trow[hh], 0.0f) * w2sh[hh];
    out[(long long)bi * Nn + j] = acc;
  }
}

// ---------------- host orchestration ----------------

extern "C" void kernel_launch(void* const* d_in, const int* in_sizes, int n_in,
                              void* d_out, int out_size, void* d_ws, size_t ws_size,
                              hipStream_t stream) {
  const float*     x   = (const float*)d_in[0];
  const float*     ew  = (const float*)d_in[1];
  const float*     Wx  = (const float*)d_in[2];
  const float*     bx  = (const float*)d_in[3];
  const float*     Wh  = (const float*)d_in[4];
  const float*     bh  = (const float*)d_in[5];
  const float*     W1  = (const float*)d_in[6];
  const float*     b1  = (const float*)d_in[7];
  const float*     W2  = (const float*)d_in[8];
  const float*     b2  = (const float*)d_in[9];
  const long long* ei  = (const long long*)d_in[10];
  float*           out = (float*)d_out;
  (void)in_sizes; (void)n_in; (void)out_size; (void)ws_size;

  // workspace carve-up (~13 MB, 4B aligned)
  char*  ws     = (char*)d_ws;
  int*   srcI   = (int*)ws;            // E
  int*   dstI   = srcI + Ee;           // E
  float* nw     = (float*)(dstI + Ee); // E
  float* deg    = nw + Ee;             // N
  int*   cnt    = (int*)(deg + Nn);    // N
  int*   fill   = cnt + Nn;            // N
  int*   rowptr = fill + Nn;           // N+1 (+pad)
  int*   srcS   = rowptr + Nn + 4;     // E (CSR-ordered src)
  float* nwS    = (float*)(srcS + Ee); // E (CSR-ordered norm weights)
  float* xt     = nwS + Ee;            // B*N*C
  float* T1x    = xt + BNC;
  float* T2x    = T1x + BNC;
  float* T1h    = T2x + BNC;           // reused for the h*r basis
  float* T2h    = T1h + BNC;
  float* preZ   = T2h + BNC;
  float* preR   = preZ + BNH;
  float* preH   = preR + BNH;
  float* hr     = preH + BNH;
  float* hbuf   = hr + BNH;
  float* sp     = hbuf + BNH;
  float* tp     = sp + BNH;

  const dim3 blk(256);
  const dim3 gE(Ee / 256);     // 64
  const dim3 gBN(BNC / 256);   // 1024
  const dim3 gGemm(128);       // 1024 waves = (4096/16)*(64/16) tiles

  // Chebyshev basis of `tin` into (t1, t2):  t1 = L@tin ; t2 = 2*L@t1 - tin
  auto basis = [&](const float* tin, float* t1, float* t2) {
    k_spmv_csr<<<gBN, blk, 0, stream>>>(t1, tin, tin, 0.0f, 1.0f, rowptr, srcS, nwS);
    k_spmv_csr<<<gBN, blk, 0, stream>>>(t2, t1, tin, -1.0f, 2.0f, rowptr, srcS, nwS);
  };

  // ---- graph normalization + CSR build (once) ----
  k_edge_prep<<<gE, blk, 0, stream>>>(ei, srcI, dstI);
  k_zero<<<dim3(6), blk, 0, stream>>>(deg, 3 * Nn + 4);  // deg, cnt, fill
  k_deg<<<gE, blk, 0, stream>>>(srcI, ew, deg);
  k_norm<<<gE, blk, 0, stream>>>(srcI, dstI, ew, deg, nw);
  k_count<<<gE, blk, 0, stream>>>(dstI, cnt);
  k_scan<<<dim3(1), dim3(1), 0, stream>>>(cnt, rowptr);
  k_scatter<<<gE, blk, 0, stream>>>(srcI, dstI, nw, rowptr, fill, srcS, nwS);

  // ---- h0 = 0 ----
  k_zero<<<gBN, blk, 0, stream>>>(hbuf, BNH);

  for (int t = 0; t < Tt; ++t) {
    // x_t basis: shared by all gates of BOTH layers (reference feeds same x_t to each layer)
    k_gather_xt<<<gBN, blk, 0, stream>>>(x, xt, t);
    basis(xt, T1x, T2x);

    for (int l = 0; l < Ll; ++l) {
      const float* Wxl = Wx + (size_t)l * 3 * Kk * Cc * Hh;
      const float* Whl = Wh + (size_t)l * 3 * Kk * Hh * Hh;
      const float* bxl = bx + (size_t)l * 3 * Hh;
      const float* bhl = bh + (size_t)l * 3 * Hh;

      // h basis: shared by z and r gates
      basis(hbuf, T1h, T2h);
      // z / r gate pre-activations (each: 6 fused matmuls + bias)
      k_gemm6<<<gGemm, blk, 0, stream>>>(xt, T1x, T2x, Wxl,
                                         hbuf, T1h, T2h, Whl,
                                         bxl, bhl, preZ);
      k_gemm6<<<gGemm, blk, 0, stream>>>(xt, T1x, T2x, Wxl + (size_t)Kk * 4096,
                                         hbuf, T1h, T2h, Whl + (size_t)Kk * 4096,
                                         bxl + 64, bhl + 64, preR);
      k_gate_zr<<<gBN, blk, 0, stream>>>(preZ, preR, hbuf, hr);

      // candidate: x basis + (h*r) basis (reuse T1h/T2h)
      basis(hr, T1h, T2h);
      k_gemm6<<<gGemm, blk, 0, stream>>>(xt, T1x, T2x, Wxl + (size_t)2 * Kk * 4096,
                                         hr, T1h, T2h, Whl + (size_t)2 * Kk * 4096,
                                         bxl + 128, bhl + 128, preH);
      k_gate_h<<<gBN, blk, 0, stream>>>(hbuf, preZ, preH);
    }
  }

  // ---- link predictor ----
  k_gemm1<<<gGemm, blk, 0, stream>>>(hbuf, W1, sp);
  k_gemm1<<<gGemm, blk, 0, stream>>>(hbuf, W1 + Hh * Hh, tp);
  k_pred<<<dim3(Bc * Nn), blk, 0, stream>>>(sp, tp, b1, W2, b2, out);
}